// EncodingShake_566935683618
// MI455X (gfx1250) — compile-verified
//
#include <hip/hip_runtime.h>

typedef __attribute__((ext_vector_type(2))) float v2f;
typedef __attribute__((ext_vector_type(8))) float v8f;

#define BATCH 16
#define DDIM  512
#define HW    3600
#define KCODE 32
#define NPIX  (BATCH * HW)   // 57600

#define TILE_PIX 256         // pixels per block in assign_kernel (8 waves x 32)
#define CT_ROW   96          // paired-codeword row stride: 96 % 64 == 32 -> conflict-free halves
#define SL_PAD   33          // padded row stride for xc staging

#define NCHUNK   48          // 3600 = 75 * 48 -> no tail, no guards
#define XROW     52          // staged X row stride (mult of 4; 52*d mod 64 distinct for d<16)

// ---------------------------------------------------------------------------
// Kernel 0: zero the per-(b,k) column-sum accumulator
// ---------------------------------------------------------------------------
__global__ void zero_asum(float* __restrict__ asum) {
    int t = blockIdx.x * blockDim.x + threadIdx.x;
    if (t < BATCH * KCODE) asum[t] = 0.0f;
}

// ---------------------------------------------------------------------------
// Kernel 1: xc GEMM (WMMA f32 16x16x4) + softmax -> A (paired layout), Asum
//   X  : (B, D, HW)  so Xf[g][d] = X[(g/HW)*D*HW + d*HW + (g%HW)]
//   A_ws paired: A[(g>>1)*64 + 2k + (g&1)]  (so aggregate reads b64 pairs)
// ---------------------------------------------------------------------------
__global__ __launch_bounds__(256) void assign_kernel(
    const float* __restrict__ X, const float* __restrict__ Cw,
    const float* __restrict__ scale, float* __restrict__ A_ws,
    float* __restrict__ Asum)
{
    __shared__ float lds_ct[(DDIM / 2) * CT_ROW];  // paired Ct[(d/2)][2k + (d&1)]
    __shared__ float lds_sl[TILE_PIX * SL_PAD];    // xc staging [pix][k]
    __shared__ float lds_x2[TILE_PIX];
    __shared__ float lds_c2[KCODE];
    __shared__ float lds_ksum[2][KCODE];

    const int t    = threadIdx.x;
    const int lane = t & 31;
    const int wave = t >> 5;

    if (t < KCODE) { lds_c2[t] = 0.0f; lds_ksum[0][t] = 0.0f; lds_ksum[1][t] = 0.0f; }
    __syncthreads();

    // Stage codewords (paired layout) into LDS; fold ||C_k||^2 into lds_c2.
    for (int i = t; i < KCODE * DDIM; i += 256) {
        int k = i >> 9;          // / DDIM
        int d = i & (DDIM - 1);
        float v = Cw[i];
        lds_ct[(d >> 1) * CT_ROW + k * 2 + (d & 1)] = v;
        atomicAdd(&lds_c2[k], v * v);
    }
    __syncthreads();

    // Each wave owns 32 consecutive pixels = two 16-pixel WMMA tiles.
    const int pixBase = blockIdx.x * TILE_PIX + wave * 32;
    const int p0 = pixBase + (lane & 15);
    const int p1 = p0 + 16;
    const int b0i = p0 / HW;
    const int b1i = p1 / HW;
    const long rb0 = (long)b0i * (DDIM * HW) + (p0 - b0i * HW);
    const long rb1 = (long)b1i * (DDIM * HW) + (p1 - b1i * HW);
    const int roff = (lane >> 4) << 1;  // lanes 0-15 -> d+{0,1}, 16-31 -> d+{2,3}
    const int kcol = lane & 15;

    v8f acc00 = {}, acc01 = {};   // tile0: k 0..15 / 16..31
    v8f acc10 = {}, acc11 = {};   // tile1
    float sq0 = 0.0f, sq1 = 0.0f;

#pragma unroll 2
    for (int d = 0; d < DDIM; d += 4) {
        const int dd = d + roff;
        float a00 = X[rb0 + (long)dd * HW];
        float a01 = X[rb0 + (long)(dd + 1) * HW];
        float a10 = X[rb1 + (long)dd * HW];
        float a11 = X[rb1 + (long)(dd + 1) * HW];
        sq0 += a00 * a00 + a01 * a01;
        sq1 += a10 * a10 + a11 * a11;
        v2f a0 = {a00, a01};
        v2f a1 = {a10, a11};

        const int cb = (dd >> 1) * CT_ROW + kcol * 2;     // 8B-aligned pair
        v2f b0 = *(const v2f*)&lds_ct[cb];                // ds_load_b64
        v2f b1 = *(const v2f*)&lds_ct[cb + 32];

        acc00 = __builtin_amdgcn_wmma_f32_16x16x4_f32(false, a0, false, b0, (short)0, acc00, false, false);
        acc01 = __builtin_amdgcn_wmma_f32_16x16x4_f32(false, a0, false, b1, (short)0, acc01, false, false);
        acc10 = __builtin_amdgcn_wmma_f32_16x16x4_f32(false, a1, false, b0, (short)0, acc10, false, false);
        acc11 = __builtin_amdgcn_wmma_f32_16x16x4_f32(false, a1, false, b1, (short)0, acc11, false, false);
    }

    // x2: lane l and l^16 hold complementary d-subsets of the same pixel.
    sq0 += __shfl_xor(sq0, 16, 32);
    sq1 += __shfl_xor(sq1, 16, 32);
    if (lane < 16) {
        lds_x2[wave * 32 + lane]      = sq0;
        lds_x2[wave * 32 + 16 + lane] = sq1;
    }

    // Dump D-tiles: value (VGPR r, lane l) is (pixel = r + 8*(l>=16), k = lane&15).
    const int rowAdd = ((lane >> 4) & 1) * 8;
#pragma unroll
    for (int r = 0; r < 8; ++r) {
        int pr0 = wave * 32 + r + rowAdd;
        int pr1 = pr0 + 16;
        lds_sl[pr0 * SL_PAD + kcol]      = acc00[r];
        lds_sl[pr0 * SL_PAD + 16 + kcol] = acc01[r];
        lds_sl[pr1 * SL_PAD + kcol]      = acc10[r];
        lds_sl[pr1 * SL_PAD + 16 + kcol] = acc11[r];
    }
    __syncthreads();

    // Softmax phase: one thread per pixel (all 256 threads active).
    {
        int g  = blockIdx.x * TILE_PIX + t;
        int bb = g / HW;
        float x2 = lds_x2[t];
        float sl[KCODE];
        float m = -3.4e38f;
#pragma unroll
        for (int k = 0; k < KCODE; ++k) {
            float v = scale[k] * (x2 - 2.0f * lds_sl[t * SL_PAD + k] + lds_c2[k]);
            sl[k] = v;
            m = fmaxf(m, v);
        }
        float s = 0.0f;
#pragma unroll
        for (int k = 0; k < KCODE; ++k) { sl[k] = __expf(sl[k] - m); s += sl[k]; }
        float inv = 1.0f / s;
        int krow = (bb == (blockIdx.x * TILE_PIX) / HW) ? 0 : 1;  // block may straddle a batch
        long pb = ((long)(g >> 1) * KCODE) * 2 + (g & 1);         // paired layout base
#pragma unroll
        for (int k = 0; k < KCODE; ++k) {
            float a = sl[k] * inv;
            A_ws[pb + 2 * k] = a;
            atomicAdd(&lds_ksum[krow][k], a);
        }
    }
    __syncthreads();

    const int bA = (blockIdx.x * TILE_PIX) / HW;
    const int bB = (blockIdx.x * TILE_PIX + TILE_PIX - 1) / HW;
    if (t < KCODE) atomicAdd(&Asum[bA * KCODE + t], lds_ksum[0][t]);
    else if (t >= 32 && t < 64 && bB != bA) atomicAdd(&Asum[bB * KCODE + (t - 32)], lds_ksum[1][t - 32]);
}

// ---------------------------------------------------------------------------
// Kernel 2: E[b,k,d] = sum_n A[b,n,k]*Xf[b,n,d] - Asum[b,k]*C[k,d]
// One block per (b, 16-wide d-tile). 8 waves split the 75 n-chunks (48 each,
// no tails). X tiles stream into per-wave double-buffered LDS via CDNA5
// async loads (global_load_async_to_lds_b128 + s_wait_asynccnt), overlapped
// with the WMMA stream. Cross-wave LDS reduction applies -Asum*C and writes E.
// ---------------------------------------------------------------------------
__global__ __launch_bounds__(256) void aggregate_kernel(
    const float* __restrict__ X, const float* __restrict__ Cw,
    const float* __restrict__ A_ws, const float* __restrict__ Asum,
    float* __restrict__ E)
{
    __shared__ __align__(16) float lds_x[8][2][16 * XROW];  // per-wave double buffer [d][n]
    __shared__ float lds_red[8][KCODE * 16];

    const int t    = threadIdx.x;
    const int lane = t & 31;
    const int wave = t >> 5;

    const int bIdx = blockIdx.x >> 5;        // 32 d-tiles per batch
    const int d0   = (blockIdx.x & 31) * 16;
    const long xbase = (long)bIdx * (DDIM * HW) + (long)d0 * HW;

    const int roff = (lane >> 4) << 1;
    const int kcol = lane & 15;

    unsigned ldsb[2];
    ldsb[0] = (unsigned)(uintptr_t)&lds_x[wave][0][0];   // low 32 bits = LDS offset
    ldsb[1] = (unsigned)(uintptr_t)&lds_x[wave][1][0];

    // Issue 6 async b128 loads covering one 16d x 48n chunk (768 floats).
    auto issue_chunk = [&](int c, int bufsel) {
        const int n0 = c * NCHUNK;
#pragma unroll
        for (int j = 0; j < 6; ++j) {
            int slot = lane + 32 * j;              // 0..191 float4 slots
            int row  = slot / 12;                  // 12 float4 per 48-float row
            int col  = (slot - row * 12) << 2;
            const float* gp = X + xbase + (long)row * HW + n0 + col;   // 16B aligned
            unsigned la = ldsb[bufsel] + (unsigned)((row * XROW + col) << 2);
            asm volatile("global_load_async_to_lds_b128 %0, %1, off"
                         :: "v"(la), "v"(gp) : "memory");
        }
    };

    v8f acc0 = {};   // k = 0..15
    v8f acc1 = {};   // k = 16..31

    const int NCH = HW / NCHUNK;                   // 75
    int buf = 0;
    issue_chunk(wave, 0);
    for (int c = wave; c < NCH; c += 8) {
        const int cn = c + 8;
        if (cn < NCH) {
            issue_chunk(cn, buf ^ 1);
            // in-order async completion: <=6 outstanding => current buffer landed
            asm volatile("s_wait_asynccnt 0x6" ::: "memory");
        } else {
            asm volatile("s_wait_asynccnt 0x0" ::: "memory");
        }
        const float* xt = &lds_x[wave][buf][0];

#pragma unroll 4
        for (int nn = 0; nn < NCHUNK; nn += 4) {
            const int n = c * NCHUNK + nn + roff;                   // even
            const long ab = ((long)(bIdx * HW + n) >> 1) * (KCODE * 2) + kcol * 2;
            v2f av0 = *(const v2f*)(A_ws + ab);                     // global b64, coalesced
            v2f av1 = *(const v2f*)(A_ws + ab + 32);
            v2f bv  = *(const v2f*)(xt + kcol * XROW + nn + roff);  // ds_load_b64

            acc0 = __builtin_amdgcn_wmma_f32_16x16x4_f32(false, av0, false, bv, (short)0, acc0, false, false);
            acc1 = __builtin_amdgcn_wmma_f32_16x16x4_f32(false, av1, false, bv, (short)0, acc1, false, false);
        }
        buf ^= 1;
    }

    // Cross-wave reduction: tile value (r, lane) is (k = r + 8*(l>=16) [+16], d = lane&15).
    const int rowAdd = ((lane >> 4) & 1) * 8;
#pragma unroll
    for (int r = 0; r < 8; ++r) {
        lds_red[wave][(r + rowAdd) * 16 + kcol]      = acc0[r];
        lds_red[wave][(16 + r + rowAdd) * 16 + kcol] = acc1[r];
    }
    __syncthreads();

    for (int idx = t; idx < KCODE * 16; idx += 256) {
        float e = 0.0f;
#pragma unroll
        for (int w = 0; w < 8; ++w) e += lds_red[w][idx];
        int k  = idx >> 4;
        int dc = idx & 15;
        e -= Asum[bIdx * KCODE + k] * Cw[k * DDIM + d0 + dc];
        E[((long)bIdx * KCODE + k) * DDIM + d0 + dc] = e;
    }
}

// ---------------------------------------------------------------------------
extern "C" void kernel_launch(void* const* d_in, const int* in_sizes, int n_in,
                              void* d_out, int out_size, void* d_ws, size_t ws_size,
                              hipStream_t stream) {
    const float* X     = (const float*)d_in[0];   // (16, 512, 60, 60)
    const float* Cw    = (const float*)d_in[1];   // (32, 512)
    const float* scale = (const float*)d_in[2];   // (32,)
    float* out  = (float*)d_out;                  // (16, 32, 512)

    float* A_ws = (float*)d_ws;                          // NPIX * KCODE floats (7.4 MB), paired layout
    float* Asum = A_ws + (size_t)NPIX * KCODE;           // BATCH * KCODE floats

    zero_asum<<<2, 256, 0, stream>>>(Asum);
    assign_kernel<<<NPIX / TILE_PIX, 256, 0, stream>>>(X, Cw, scale, A_ws, Asum);
    aggregate_kernel<<<BATCH * 32, 256, 0, stream>>>(X, Cw, A_ws, Asum, out);
}